// LossComputation_5102421147884
// MI455X (gfx1250) — compile-verified
//
#include <hip/hip_runtime.h>

typedef __attribute__((ext_vector_type(2))) float v2f;
typedef __attribute__((ext_vector_type(8))) float v8f;

#define B_SZ   4096
#define D_SZ   512
#define C_SZ   5000
#define LSCALE 28.0f
#define ALPHA  0.6f
#define BETA   0.4f
#define SPOS   10.0f
#define SNEG   40.0f
#define NTILE_C 313          // ceil(5000/16)
#define AS_STRIDE 516        // 512 + 4 pad: rows land on distinct LDS banks
#define NEG_BIG (-3.0e38f)

// ---------- per-column inverse norm of W [D, C] over dim 0 ----------
__global__ __launch_bounds__(256) void colnorm_kernel(const float* __restrict__ W,
                                                      float* __restrict__ inv_norm) {
  int c = blockIdx.x * blockDim.x + threadIdx.x;
  if (c >= C_SZ) return;
  float s = 0.f;
  for (int d = 0; d < D_SZ; ++d) { float w = W[d * C_SZ + c]; s += w * w; }
  inv_norm[c] = rsqrtf(s);
}

// ---------- fused GEMM + online log-sum-exp cross entropy ----------
// grid = (256 row-tiles, 2 embeds), block = 128 threads = 4 waves.
__global__ __launch_bounds__(128) void ce_kernel(const float* __restrict__ V,
                                                 const float* __restrict__ T,
                                                 const float* __restrict__ W,
                                                 const int*   __restrict__ labels,
                                                 const float* __restrict__ inv_norm,
                                                 float* __restrict__ ce_part) {
  __shared__ float As[16 * AS_STRIDE];
  __shared__ float lm[4][16], ls[4][16], lxl[4][16], lnll[16];

  const float* E   = blockIdx.y ? T : V;
  const int tid    = threadIdx.x;
  const int lane   = tid & 31;
  const int wave   = tid >> 5;
  const int r0     = blockIdx.x * 16;
  const int mlo    = lane & 15;      // A row / B,C column within tile
  const int khalf  = lane >> 4;      // 0 or 1
  const int koff   = 2 * khalf;      // K sub-offset per fp32 WMMA layout

  // stage the 16x512 A tile into LDS (rows of E are contiguous -> coalesced)
  for (int i = tid; i < 16 * D_SZ; i += 128) {
    int rr = i >> 9, cc = i & 511;
    As[rr * AS_STRIDE + cc] = E[(r0 + rr) * D_SZ + cc];
  }
  __syncthreads();

  int lbl[8];
  for (int r = 0; r < 8; ++r) lbl[r] = labels[r0 + r + 8 * khalf];

  float m[8], s[8], xl[8];
  for (int r = 0; r < 8; ++r) { m[r] = NEG_BIG; s[r] = 0.f; xl[r] = NEG_BIG; }

  const float* arow = As + mlo * AS_STRIDE + koff;

  for (int jt = wave; jt < NTILE_C; jt += 4) {   // wave-uniform loop
    int col    = jt * 16 + mlo;
    int colc   = col < C_SZ ? col : (C_SZ - 1);
    bool valid = col < C_SZ;
    float invn = inv_norm[colc];

    v8f acc = {};
    for (int kk = 0; kk < D_SZ; kk += 4) {
      v2f a; a.x = arow[kk]; a.y = arow[kk + 1];
      int bk = kk + koff;
      v2f b; b.x = W[bk * C_SZ + colc]; b.y = W[(bk + 1) * C_SZ + colc];
      acc = __builtin_amdgcn_wmma_f32_16x16x4_f32(false, a, false, b,
                                                  (short)0, acc, false, false);
    }
    // branchless online softmax update (hardware v_exp_f32, no libm branches)
    for (int r = 0; r < 8; ++r) {
      float x   = LSCALE * acc[r] * invn;
      float xs  = valid ? x : NEG_BIG;             // masked value for the max
      float nm  = fmaxf(m[r], xs);
      float add = valid ? __expf(x - nm) : 0.f;    // masked contribution
      s[r] = s[r] * __expf(m[r] - nm) + add;       // s==0 while m is sentinel
      m[r] = nm;
      xl[r] = (valid && col == lbl[r]) ? x : xl[r];
    }
  }

  // merge (m,s,xl) across the 16 lanes of each half-wave (column-sharded over lanes)
  for (int off = 1; off < 16; off <<= 1) {
    for (int r = 0; r < 8; ++r) {
      float om = __shfl_xor(m[r],  off, 32);
      float os = __shfl_xor(s[r],  off, 32);
      float ox = __shfl_xor(xl[r], off, 32);
      float nm = fmaxf(m[r], om);
      s[r] = s[r] * __expf(m[r] - nm) + os * __expf(om - nm);
      m[r] = nm;
      xl[r] = fmaxf(xl[r], ox);
    }
  }
  if (mlo == 0) {                 // lanes 0 and 16 hold reduced rows
    for (int r = 0; r < 8; ++r) {
      int row = r + 8 * khalf;
      lm[wave][row] = m[r]; ls[wave][row] = s[r]; lxl[wave][row] = xl[r];
    }
  }
  __syncthreads();

  if (tid < 16) {                 // merge the 4 waves' column shards per row
    float M = NEG_BIG, S = 0.f, X = NEG_BIG;
    for (int w = 0; w < 4; ++w) {
      float om = lm[w][tid], os = ls[w][tid], ox = lxl[w][tid];
      float nm = fmaxf(M, om);
      S = S * __expf(M - nm) + os * __expf(om - nm);
      M = nm;
      X = fmaxf(X, ox);
    }
    lnll[tid] = (M + __logf(S)) - X;   // nll = lse - label_logit
  }
  __syncthreads();
  if (tid == 0) {
    float t = 0.f;
    for (int i = 0; i < 16; ++i) t += lnll[i];
    ce_part[blockIdx.y * 256 + blockIdx.x] = t;
  }
}

// ---------- sim = V @ T^T fused with softplus pair loss ----------
// grid = (256, 256) 16x16 tiles, one wave per tile.
__global__ __launch_bounds__(32) void sim_kernel(const float* __restrict__ V,
                                                 const float* __restrict__ T,
                                                 const int*   __restrict__ labels,
                                                 float* __restrict__ sim_part) {
  const int lane  = threadIdx.x;
  const int mlo   = lane & 15;
  const int khalf = lane >> 4;
  const int koff  = 2 * khalf;
  const int it = blockIdx.y, jt = blockIdx.x;
  const float* aptr = V + (size_t)(it * 16 + mlo) * D_SZ + koff;
  const float* bptr = T + (size_t)(jt * 16 + mlo) * D_SZ + koff;  // B[k][n] = T[n][k]

  v8f acc = {};
  for (int kk = 0; kk < D_SZ; kk += 4) {
    v2f a = *(const v2f*)(aptr + kk);
    v2f b = *(const v2f*)(bptr + kk);
    acc = __builtin_amdgcn_wmma_f32_16x16x4_f32(false, a, false, b,
                                                (short)0, acc, false, false);
  }

  int col  = jt * 16 + mlo;
  int lblc = labels[col];
  float part = 0.f;
  for (int r = 0; r < 8; ++r) {
    int row  = it * 16 + r + 8 * khalf;
    int lblr = labels[row];
    float x  = acc[r];
    float z  = (lblr == lblc) ? (-SPOS * (x - ALPHA)) : (SNEG * (x - BETA));
    // stable softplus with hardware transcendentals: arg of log is in (1,2]
    part += fmaxf(z, 0.f) + __logf(1.f + __expf(-fabsf(z)));
  }
  for (int off = 16; off >= 1; off >>= 1) part += __shfl_xor(part, off, 32);
  if (lane == 0) sim_part[it * 256 + jt] = part;
}

// ---------- deterministic final reduction + output ----------
__global__ __launch_bounds__(256) void reduce_kernel(const float* __restrict__ ce_part,
                                                     const float* __restrict__ sim_part,
                                                     float* __restrict__ out) {
  __shared__ float sv[256], st[256], sp[256];
  int t = threadIdx.x;
  float pv = ce_part[t];
  float pt = ce_part[256 + t];
  float ps = 0.f;
  for (int k = 0; k < 256; ++k) ps += sim_part[t + 256 * k];  // fixed order
  sv[t] = pv; st[t] = pt; sp[t] = ps;
  __syncthreads();
  for (int off = 128; off >= 1; off >>= 1) {
    if (t < off) { sv[t] += sv[t + off]; st[t] += st[t + off]; sp[t] += sp[t + off]; }
    __syncthreads();
  }
  if (t == 0) {
    float loss_v = sv[0] / (float)B_SZ;
    float loss_t = st[0] / (float)B_SZ;
    out[0] = loss_v + loss_t;               // instance_loss
    out[1] = 2.f * sp[0] / (float)B_SZ;     // global_align_loss
    out[2] = loss_v;
    out[3] = loss_t;
  }
}

extern "C" void kernel_launch(void* const* d_in, const int* in_sizes, int n_in,
                              void* d_out, int out_size, void* d_ws, size_t ws_size,
                              hipStream_t stream) {
  const float* V      = (const float*)d_in[0];
  const float* T      = (const float*)d_in[1];
  const int*   labels = (const int*)d_in[2];
  const float* W      = (const float*)d_in[3];
  float* out = (float*)d_out;
  float* ws  = (float*)d_ws;

  float* inv_norm = ws;              // 5000 floats
  float* ce_part  = ws + 5000;       // 512 floats  (256 v-tiles + 256 t-tiles)
  float* sim_part = ws + 5512;       // 65536 floats (256x256 tiles)

  colnorm_kernel<<<dim3((C_SZ + 255) / 256), dim3(256), 0, stream>>>(W, inv_norm);
  ce_kernel<<<dim3(256, 2), dim3(128), 0, stream>>>(V, T, W, labels, inv_norm, ce_part);
  sim_kernel<<<dim3(256, 256), dim3(32), 0, stream>>>(V, T, labels, sim_part);
  reduce_kernel<<<dim3(1), dim3(256), 0, stream>>>(ce_part, sim_part, out);
}